// SparseSupervisedTree_4947802325041
// MI455X (gfx1250) — compile-verified
//
#include <hip/hip_runtime.h>

// Problem constants (from reference)
#define N_INNER 2000
#define N_LEAF  10000
#define N_DOC   128
#define NM_K    2048      // node_mass K padded to 2048 (zeros), even #chunks
#define KC      128       // K chunk for WMMA GEMM
#define BST     132       // padded LDS stride for transposed B tile (even, bank-spread)
#define CH      128       // feature chunk for distance kernel
#define CHS     (CH + 1)  // padded LDS stride (odd -> conflict free)

typedef __attribute__((ext_vector_type(2))) float v2f;
typedef __attribute__((ext_vector_type(8))) float v8f;

// ---------------------------------------------------------------------------
// Stage 1a: segmented column argmax of param (2000 x 10000).
// grid = (40, 8): 8 row-segments of 250 rows for wave occupancy (2560 waves),
// fully coalesced row reads (256 contiguous floats per block per row).
// ---------------------------------------------------------------------------
__global__ void k_argmax_part(const float* __restrict__ param,
                              float* __restrict__ pmax, int* __restrict__ pidx) {
    int col = blockIdx.x * 256 + threadIdx.x;
    int seg = blockIdx.y;
    if (col >= N_LEAF) return;
    int r0 = seg * 250;
    float vmax = param[(size_t)r0 * N_LEAF + col];
    int   imax = r0;
    #pragma unroll 5
    for (int r = r0 + 1; r < r0 + 250; ++r) {
        float v = param[(size_t)r * N_LEAF + col];
        if (v > vmax) { vmax = v; imax = r; }   // strict > keeps smallest row
    }
    pmax[seg * N_LEAF + col] = vmax;
    pidx[seg * N_LEAF + col] = imax;
}

// Stage 1b: combine the 8 segment partials (ascending scan + strict > keeps
// the smallest argmax row, matching jax semantics for non-tied data).
__global__ void k_argmax_comb(const float* __restrict__ pmax,
                              const int* __restrict__ pidx,
                              int* __restrict__ amax) {
    int col = blockIdx.x * 256 + threadIdx.x;
    if (col >= N_LEAF) return;
    float vmax = pmax[col];
    int   imax = pidx[col];
    #pragma unroll
    for (int s = 1; s < 8; ++s) {
        float v = pmax[s * N_LEAF + col];
        if (v > vmax) { vmax = v; imax = pidx[s * N_LEAF + col]; }
    }
    amax[col] = imax;
}

// ---------------------------------------------------------------------------
// Stage 2: node_mass[d][j] = sum_{l: amax[l]==j} mass[d][l].
// One block per doc; accumulate in LDS (ds_add_f32), write full padded row
// (so no global zero-init pass is needed).
// ---------------------------------------------------------------------------
__global__ void k_scatter(const float* __restrict__ mass,
                          const int* __restrict__ amax,
                          float* __restrict__ node_mass) {
    __shared__ float nm[NM_K];
    int d = blockIdx.x, t = threadIdx.x;
    for (int i = t; i < NM_K; i += 256) nm[i] = 0.f;
    __syncthreads();
    const float* mrow = mass + (size_t)d * N_LEAF;
    for (int l = t; l < N_LEAF; l += 256)
        atomicAdd(&nm[amax[l]], mrow[l]);            // ds_add_f32
    __syncthreads();
    float* orow = node_mass + (size_t)d * NM_K;
    for (int i = t; i < NM_K; i += 256) orow[i] = nm[i];
}

// ---------------------------------------------------------------------------
// Stage 3: proj(128 x 2000) = node_mass(128 x 2048) @ S^T via
// V_WMMA_F32_16X16X4_F32.  B[k][n] = S[n0+n][k] (subtree indicator) is
// generated on the fly in LDS with the ancestor walk p = (p-1)/5.
// grid = 125 N-tiles, block = 8 waves; wave w owns docs 16w..16w+15.
// A layout (16x4 f32): lanes 0-15 hold K=0(v0),K=1(v1); lanes 16-31 K=2,K=3.
// B layout mirrored on K; C/D: VGPR r = rows r (lanes 0-15) and r+8 (16-31).
// ---------------------------------------------------------------------------
__global__ void k_proj_wmma(const float* __restrict__ node_mass,
                            float* __restrict__ proj) {
    __shared__ float BsT[16 * BST];   // [n][k] transposed, stride 132
    const int n0   = blockIdx.x * 16;
    const int t    = threadIdx.x;
    const int wave = t >> 5;
    const int lane = t & 31;
    const int nl   = lane & 15;       // row (A) / col (B) within tile
    const int half = lane >> 4;       // selects K pair {0,1} vs {2,3}

    v8f acc = {};
    for (int c = 0; c < NM_K / KC; ++c) {
        __syncthreads();              // previous tile fully consumed
        // Build B chunk: 128 k-values x 16 n-values, 8 entries per thread.
        #pragma unroll
        for (int e = 0; e < (KC * 16) / 256; ++e) {
            int flat = e * 256 + t;
            int kk = flat >> 4, nn = flat & 15;
            int k = c * KC + kk, n = n0 + nn;
            float s = 0.f;
            if (k < N_INNER) {
                int p = k;
                while (p > n) p = (p - 1) / 5;   // parent walk (<=6 steps)
                s = (p == n) ? 1.f : 0.f;
            }
            BsT[nn * BST + kk] = s;
        }
        __syncthreads();
        const float* arow =
            node_mass + (size_t)(wave * 16 + nl) * NM_K + c * KC + 2 * half;
        const float* brow = &BsT[nl * BST + 2 * half];
        #pragma unroll 8
        for (int kk = 0; kk < KC; kk += 4) {
            v2f a = *(const v2f*)(arow + kk);     // global_load_b64
            v2f b = *(const v2f*)(brow + kk);     // ds_load_b64
            acc = __builtin_amdgcn_wmma_f32_16x16x4_f32(
                false, a, false, b, (short)0, acc, false, false);
        }
    }
    // Scatter D: element r -> (doc = 16w + r + 8*half, col = n0 + nl)
    #pragma unroll
    for (int r = 0; r < 8; ++r) {
        int doc = wave * 16 + r + 8 * half;
        proj[(size_t)doc * N_INNER + n0 + nl] = acc[r];
    }
}

// ---------------------------------------------------------------------------
// Stage 4: out[i][j] = |proj_i - proj_j|_1 + |mass_i - mass_j|_1.
// 16x16 doc-pair tile per block, features staged through LDS in 128-chunks.
// ---------------------------------------------------------------------------
__global__ void k_dist(const float* __restrict__ mass,
                       const float* __restrict__ proj,
                       float* __restrict__ out) {
    __shared__ float As[16 * CHS];
    __shared__ float Bs[16 * CHS];
    const int ti = threadIdx.x & 15, tj = threadIdx.x >> 4;
    const int i0 = blockIdx.x * 16, j0 = blockIdx.y * 16;
    float acc = 0.f;
    #pragma unroll
    for (int part = 0; part < 2; ++part) {
        const float* base = part ? mass : proj;
        const int len = part ? N_LEAF : N_INNER;
        for (int k0 = 0; k0 < len; k0 += CH) {
            __syncthreads();
            #pragma unroll
            for (int e = 0; e < (16 * CH) / 256; ++e) {
                int flat = e * 256 + threadIdx.x;
                int row = flat / CH, cc = flat % CH;
                int k = k0 + cc;
                float av = 0.f, bv = 0.f;
                if (k < len) {
                    av = base[(size_t)(i0 + row) * len + k];
                    bv = base[(size_t)(j0 + row) * len + k];
                }
                As[row * CHS + cc] = av;
                Bs[row * CHS + cc] = bv;
            }
            __syncthreads();
            #pragma unroll 8
            for (int kk = 0; kk < CH; ++kk)
                acc += fabsf(As[ti * CHS + kk] - Bs[tj * CHS + kk]);
        }
    }
    out[(size_t)(i0 + ti) * N_DOC + (j0 + tj)] = acc;
}

// ---------------------------------------------------------------------------
// Workspace layout (bytes), total ~2.75 MB:
//   pmax      @ 0        : 8*10000 f32  (320000)
//   pidx      @ 320000   : 8*10000 i32  (320000)
//   amax      @ 640000   : 10000  i32   (40000)
//   node_mass @ 680000   : 128*2048 f32 (1048576)   (8-byte aligned)
//   proj      @ 1728576  : 128*2000 f32 (1024000)   (8-byte aligned)
// ---------------------------------------------------------------------------
extern "C" void kernel_launch(void* const* d_in, const int* in_sizes, int n_in,
                              void* d_out, int out_size, void* d_ws, size_t ws_size,
                              hipStream_t stream) {
    const float* mass  = (const float*)d_in[0];   // (128, 10000)
    const float* param = (const float*)d_in[1];   // (2000, 10000)
    float* out = (float*)d_out;                   // (128, 128)
    char* ws = (char*)d_ws;

    float* pmax      = (float*)(ws);
    int*   pidx      = (int*)  (ws + 320000);
    int*   amax      = (int*)  (ws + 640000);
    float* node_mass = (float*)(ws + 680000);
    float* proj      = (float*)(ws + 1728576);

    k_argmax_part<<<dim3(40, 8), 256, 0, stream>>>(param, pmax, pidx);
    k_argmax_comb<<<40, 256, 0, stream>>>(pmax, pidx, amax);
    k_scatter<<<N_DOC, 256, 0, stream>>>(mass, amax, node_mass);
    k_proj_wmma<<<N_INNER / 16, 256, 0, stream>>>(node_mass, proj);
    k_dist<<<dim3(8, 8), 256, 0, stream>>>(mass, proj, out);
}